// ArcFaceLoss_12927851561776
// MI455X (gfx1250) — compile-verified
//
#include <hip/hip_runtime.h>
#include <math.h>

typedef __attribute__((ext_vector_type(2))) float v2f;
typedef __attribute__((ext_vector_type(8))) float v8f;
typedef __attribute__((__vector_size__(16))) int i32x4;   // matches builtin param type

#define B_ROWS 1024
#define D_DIM  512
#define C_COLS 100000
#define S_SCALE 64.0f
#define MARGIN  0.1f
#define EPSN    1e-12f

#define CN   160              // columns per block (100000 = 160 * 625)
#define NT   10               // 16-col WMMA tiles per wave (160/16)
#define RM   128              // rows per row-group (8 waves * 16 rows)
#define KT   32               // K tile staged in LDS
#define KTP  36               // padded row stride: 36*i mod 64 distinct for 16 rows, 16B-aligned
#define NKT  (D_DIM / KT)     // 16 K-tiles
#define NBLK (C_COLS / CN)    // 625 blocks

#if __has_builtin(__builtin_amdgcn_global_load_async_to_lds_b128) && \
    __has_builtin(__builtin_amdgcn_s_wait_asynccnt)
#define HAVE_ASYNC 1
#else
#define HAVE_ASYNC 0
#endif

#define GAS __attribute__((address_space(1)))
#define LAS __attribute__((address_space(3)))

// ---------------- kernel 1: normalize input rows (x / max(||x||, eps)) ----------
__global__ __launch_bounds__(256) void k_norm_inputs(const float* __restrict__ x,
                                                     float* __restrict__ xn) {
  int wave = (blockIdx.x * blockDim.x + threadIdx.x) >> 5;
  int lane = threadIdx.x & 31;
  if (wave >= B_ROWS) return;
  const float* row = x + (size_t)wave * D_DIM;
  float ss = 0.f;
#pragma unroll
  for (int i = 0; i < D_DIM / 32; ++i) { float v = row[lane + 32 * i]; ss += v * v; }
#pragma unroll
  for (int m = 16; m >= 1; m >>= 1) ss += __shfl_xor(ss, m, 32);
  float inv = 1.0f / fmaxf(sqrtf(ss), EPSN);
  float* o = xn + (size_t)wave * D_DIM;
#pragma unroll
  for (int i = 0; i < D_DIM / 32; ++i) o[lane + 32 * i] = row[lane + 32 * i] * inv;
}

// ---------------- kernel 2: per-row 1/max(||w||,eps) ----------------------------
__global__ __launch_bounds__(256) void k_weight_invnorm(const float* __restrict__ w,
                                                        float* __restrict__ invw) {
  int wave = (blockIdx.x * blockDim.x + threadIdx.x) >> 5;
  int lane = threadIdx.x & 31;
  if (wave >= C_COLS) return;
  const float* row = w + (size_t)wave * D_DIM;
  float ss = 0.f;
#pragma unroll
  for (int i = 0; i < D_DIM / 32; ++i) { float v = row[lane + 32 * i]; ss += v * v; }
#pragma unroll
  for (int m = 16; m >= 1; m >>= 1) ss += __shfl_xor(ss, m, 32);
  if (lane == 0) invw[wave] = 1.0f / fmaxf(sqrtf(ss), EPSN);
}

// ---------------- GEMM helpers --------------------------------------------------
__device__ __forceinline__ void load_tile(int tid, const float* __restrict__ xn,
                                          const float* __restrict__ weight,
                                          int c0, int rg, int kt,
                                          float* Asb, float* Bsb) {
#if HAVE_ASYNC
#pragma unroll
  for (int i = 0; i < 4; ++i) {            // A: 128x32 = 1024 B128 lanes, 4 instrs/wave
    int q = tid + 256 * i;
    int r = q >> 3, kq = (q & 7) << 2;
    __builtin_amdgcn_global_load_async_to_lds_b128(
        (GAS i32x4*)(xn + (size_t)(rg * RM + r) * D_DIM + kt + kq),
        (LAS i32x4*)(Asb + r * KTP + kq), 0, 0);
  }
#pragma unroll
  for (int i = 0; i < 5; ++i) {            // B: 160x32 = 1280 B128 lanes, 5 instrs/wave
    int q = tid + 256 * i;
    int r = q >> 3, kq = (q & 7) << 2;
    __builtin_amdgcn_global_load_async_to_lds_b128(
        (GAS i32x4*)(weight + (size_t)(c0 + r) * D_DIM + kt + kq),
        (LAS i32x4*)(Bsb + r * KTP + kq), 0, 0);
  }
#else
#pragma unroll
  for (int i = 0; i < 4; ++i) {
    int q = tid + 256 * i;
    int r = q >> 3, kq = (q & 7) << 2;
    float4 v = *(const float4*)(xn + (size_t)(rg * RM + r) * D_DIM + kt + kq);
    *(float4*)(&Asb[r * KTP + kq]) = v;
  }
#pragma unroll
  for (int i = 0; i < 5; ++i) {
    int q = tid + 256 * i;
    int r = q >> 3, kq = (q & 7) << 2;
    float4 v = *(const float4*)(weight + (size_t)(c0 + r) * D_DIM + kt + kq);
    *(float4*)(&Bsb[r * KTP + kq]) = v;
  }
#endif
}

__device__ __forceinline__ void compute_tile(const float* Asb, const float* Bsb,
                                             int wave, int m, int kk, v8f (&acc)[NT]) {
#pragma unroll
  for (int k4 = 0; k4 < KT / 4; ++k4) {
    const int kb = k4 * 4 + kk;
    v2f a;
    a.x = Asb[(wave * 16 + m) * KTP + kb];
    a.y = Asb[(wave * 16 + m) * KTP + kb + 1];
#pragma unroll
    for (int t = 0; t < NT; ++t) {
      v2f b;
      b.x = Bsb[(16 * t + m) * KTP + kb];
      b.y = Bsb[(16 * t + m) * KTP + kb + 1];
      acc[t] = __builtin_amdgcn_wmma_f32_16x16x4_f32(
          false, a, false, b, (short)0, acc[t], false, false);
    }
  }
}

// ---------------- kernel 3: fused WMMA GEMM + ArcFace logits + exp-sum ----------
__global__ __launch_bounds__(256) void k_arcface_gemm(
    const float* __restrict__ xn, const float* __restrict__ weight,
    const float* __restrict__ invw, const int* __restrict__ labels,
    float* __restrict__ lab_logit, float* __restrict__ partial) {
  __shared__ float As[2][RM * KTP];   // double-buffered 128 x 32 (padded)
  __shared__ float Bs[2][CN * KTP];   // double-buffered 160 x 32 (padded)

  const int tid   = threadIdx.x;
  const int lane  = tid & 31;
  const int wave  = tid >> 5;
  const int c0    = blockIdx.x * CN;
  const int m     = lane & 15;               // col in 16-tile / A-frag row
  const int kk    = (lane < 16) ? 0 : 2;     // K sub-offset per ISA 32-bit A/B layout
  const int m_off = (lane < 16) ? 0 : 8;     // C/D: lanes 16-31 hold M = v+8

  float invc[NT];
#pragma unroll
  for (int t = 0; t < NT; ++t) invc[t] = invw[c0 + 16 * t + m];

  for (int rg = 0; rg < B_ROWS / RM; ++rg) {
    v8f acc[NT];
#pragma unroll
    for (int t = 0; t < NT; ++t) acc[t] = (v8f){0.f,0.f,0.f,0.f,0.f,0.f,0.f,0.f};

    // -------- double-buffered K pipeline: tile k+1 streams while tile k computes
    load_tile(tid, xn, weight, c0, rg, 0, As[0], Bs[0]);
#pragma unroll 1
    for (int kt = 0; kt < NKT - 1; ++kt) {
      load_tile(tid, xn, weight, c0, rg, (kt + 1) * KT, As[(kt + 1) & 1], Bs[(kt + 1) & 1]);
#if HAVE_ASYNC
      __builtin_amdgcn_s_wait_asynccnt(9);   // tile kt done; tile kt+1 (9 instrs) in flight
#endif
      __syncthreads();
      compute_tile(As[kt & 1], Bs[kt & 1], wave, m, kk, acc);
      __syncthreads();
    }
#if HAVE_ASYNC
    __builtin_amdgcn_s_wait_asynccnt(0);
#endif
    __syncthreads();
    compute_tile(As[(NKT - 1) & 1], Bs[(NKT - 1) & 1], wave, m, kk, acc);
    __syncthreads();

    // ---- epilogue: cos -> logits (margin at label col) -> exp-sum (fixed max = S)
    const int r_base = rg * RM + wave * 16;
    float rs[8];
#pragma unroll
    for (int v = 0; v < 8; ++v) rs[v] = 0.f;

#pragma unroll
    for (int v = 0; v < 8; ++v) {
      const int row = r_base + m_off + v;
      const int lb  = labels[row];
#pragma unroll
      for (int t = 0; t < NT; ++t) {
        const int col = c0 + 16 * t + m;
        float cosv  = fminf(fmaxf(acc[t][v] * invc[t], -1.0f), 1.0f);
        float logit = S_SCALE * cosv;
        if (lb == col) {
          float labv = S_SCALE * cosf(acosf(cosv) + MARGIN);
          lab_logit[row] = labv;     // exactly one block owns each row's label col
          logit = labv;
        }
        rs[v] += __expf(logit - S_SCALE);
      }
    }
    // reduce over the 16 lanes sharing each row-half (xor masks < 16 stay in-half)
#pragma unroll
    for (int v = 0; v < 8; ++v) {
#pragma unroll
      for (int mm = 8; mm >= 1; mm >>= 1) rs[v] += __shfl_xor(rs[v], mm, 32);
    }
    if (m == 0) {  // lanes 0 (rows v) and 16 (rows v+8) write deterministically
#pragma unroll
      for (int v = 0; v < 8; ++v)
        partial[(size_t)(r_base + m_off + v) * NBLK + blockIdx.x] = rs[v];
    }
  }
}

// ---------------- kernel 4: deterministic reduce + loss ------------------------
__global__ __launch_bounds__(256) void k_finalize(const float* __restrict__ partial,
                                                  const float* __restrict__ lab_logit,
                                                  float* __restrict__ out) {
  __shared__ float red[256];
  int tid = threadIdx.x;
  float lsum = 0.f;
  for (int r = tid; r < B_ROWS; r += 256) {
    float s = 0.f;
    for (int j = 0; j < NBLK; ++j) s += partial[(size_t)r * NBLK + j];
    // loss_r = -(lab - logsumexp) ; logsumexp = log(sum exp(l - S)) + S
    lsum += (logf(s) + S_SCALE) - lab_logit[r];
  }
  red[tid] = lsum;
  __syncthreads();
  for (int st = 128; st >= 1; st >>= 1) {
    if (tid < st) red[tid] += red[tid + st];
    __syncthreads();
  }
  if (tid == 0) out[0] = red[0] / (float)B_ROWS;
}

// ---------------- launcher ------------------------------------------------------
extern "C" void kernel_launch(void* const* d_in, const int* in_sizes, int n_in,
                              void* d_out, int out_size, void* d_ws, size_t ws_size,
                              hipStream_t stream) {
  const float* x      = (const float*)d_in[0];   // [1024,512]
  const int*   labels = (const int*)d_in[1];     // [1024]
  const float* w      = (const float*)d_in[2];   // [100000,512]
  float* out = (float*)d_out;

  float* ws  = (float*)d_ws;
  float* xn      = ws;                            // 1024*512   = 524288
  float* invw    = xn + (size_t)B_ROWS * D_DIM;   // 100000
  float* labl    = invw + C_COLS;                 // 1024
  float* partial = labl + B_ROWS;                 // 1024*625   = 640000  (~4.8 MB total)

  k_norm_inputs   <<<B_ROWS / 8, 256, 0, stream>>>(x, xn);
  k_weight_invnorm<<<C_COLS / 8, 256, 0, stream>>>(w, invw);
  k_arcface_gemm  <<<NBLK,       256, 0, stream>>>(xn, w, invw, labels, labl, partial);
  k_finalize      <<<1,          256, 0, stream>>>(partial, labl, out);
}